// GPNN_CAD_15839839387969
// MI455X (gfx1250) — compile-verified
//
#include <hip/hip_runtime.h>
#include <hip/hip_bf16.h>

// ---------------------------------------------------------------------------
// GPNN-CAD forward for MI455X (gfx1250), fp32 via V_WMMA_F32_16X16X4_F32,
// global->LDS staging through the gfx1250 async-to-LDS path (ASYNCcnt).
// ---------------------------------------------------------------------------

typedef float v2f __attribute__((ext_vector_type(2)));
typedef float v8f __attribute__((ext_vector_type(8)));

#define GB 4
#define GT 16
#define MAXN 13
#define IN_DIM 25600
#define NF 64
#define C7 7
#define BT (GB * GT)            // 64
#define NODES (BT * MAXN)       // 832
#define PIX (BT * MAXN * MAXN)  // 10816
#define PIX_TILES (PIX / 16)    // 676
#define KSPLIT 16
#define KCHUNK (IN_DIM / KSPLIT) // 1600

__device__ __forceinline__ float sigm(float x) { return 1.0f / (1.0f + __expf(-x)); }

__device__ __forceinline__ v8f wmma4(v2f a, v2f b, v8f c) {
    return __builtin_amdgcn_wmma_f32_16x16x4_f32(false, a, false, b, (short)0, c, false, false);
}

// ---- async global->LDS 16B copy (gfx1250 GLOBAL_LOAD_ASYNC_TO_LDS_B128) ----
#if defined(__gfx1250__) && __has_builtin(__builtin_amdgcn_global_load_async_to_lds_b128)
#define ASYNC_LDS 1
typedef int v4i __attribute__((ext_vector_type(4)));
typedef __attribute__((address_space(1))) v4i* as1v4;
typedef __attribute__((address_space(3))) v4i* as3v4;
__device__ __forceinline__ void g2l16(const float* g, float* l) {
    __builtin_amdgcn_global_load_async_to_lds_b128((as1v4)g, (as3v4)l, 0, 0);
}
__device__ __forceinline__ void g2l_wait() {
#if __has_builtin(__builtin_amdgcn_s_wait_asynccnt)
    __builtin_amdgcn_s_wait_asynccnt(0);
#else
    asm volatile("s_wait_asynccnt 0" ::: "memory");
#endif
}
#else
#define ASYNC_LDS 0
__device__ __forceinline__ void g2l16(const float* g, float* l) {
    float4 v = *(const float4*)g;
    l[0] = v.x; l[1] = v.y; l[2] = v.z; l[3] = v.w;
}
__device__ __forceinline__ void g2l_wait() {}
#endif

// ---------------------------------------------------------------------------
// Kernel 0: node <- broadcast bias (K-split partials accumulate on top);
//           msum <- 0
// ---------------------------------------------------------------------------
__global__ void init_kernel(float* __restrict__ node, float* __restrict__ msum,
                            const float* __restrict__ bias) {
    int i = blockIdx.x * blockDim.x + threadIdx.x;
    if (i < NODES * NF) node[i] = bias[i & (NF - 1)];
    if (i < NODES * 128) msum[i] = 0.0f;
}

// ---------------------------------------------------------------------------
// Kernel 1: node += pose(832,25600) @ W_fc.T(25600,64), K split 16 ways.
// grid (52, 16) x 128 threads (4 waves). Wave = one 16x16 C tile of one K chunk.
// ---------------------------------------------------------------------------
#define KC 64
__global__ __launch_bounds__(128) void node_fc_kernel(
    const float* __restrict__ X,    // (832, 25600)
    const float* __restrict__ W,    // (64, 25600)
    float* __restrict__ node)       // (832, 64) pre-seeded with bias
{
    __shared__ float Al[16][KC + 4];   // stride 68: 16B-aligned rows, conflict-free cols
    __shared__ float Bl[64][KC + 4];
    const int tid  = threadIdx.x;
    const int wave = tid >> 5;
    const int lane = tid & 31;
    const int row0 = blockIdx.x * 16;
    const int col0 = wave * 16;
    const int kc0  = blockIdx.y * KCHUNK;
    const int lm = lane & 15;
    const int kh = (lane >> 4) << 1;  // 0 or 2
    const int mh = (lane >> 4) << 3;  // 0 or 8

    v8f acc = {};
    for (int kc = kc0; kc < kc0 + KCHUNK; kc += KC) {
        #pragma unroll
        for (int i = 0; i < 2; ++i) {               // A: 16 x 64 (256 float4)
            int f  = tid + i * 128;
            int r  = f >> 4;
            int kq = (f & 15) << 2;
            g2l16(X + (size_t)(row0 + r) * IN_DIM + kc + kq, &Al[r][kq]);
        }
        #pragma unroll
        for (int i = 0; i < 8; ++i) {               // B: 64 x 64 (1024 float4)
            int f  = tid + i * 128;
            int r  = f >> 4;
            int kq = (f & 15) << 2;
            g2l16(W + (size_t)r * IN_DIM + kc + kq, &Bl[r][kq]);
        }
        if (kc + KC < kc0 + KCHUNK) {
            __builtin_prefetch(X + (size_t)(row0 + (tid & 15)) * IN_DIM + kc + KC, 0, 0);
        }
        g2l_wait();
        __syncthreads();
        #pragma unroll 4
        for (int kk = 0; kk < KC; kk += 4) {
            v2f a, b;
            a.x = Al[lm][kk + kh];
            a.y = Al[lm][kk + kh + 1];
            b.x = Bl[col0 + lm][kk + kh];
            b.y = Bl[col0 + lm][kk + kh + 1];
            acc = wmma4(a, b, acc);
        }
        __syncthreads();
    }
    #pragma unroll
    for (int v = 0; v < 8; ++v) {
        int m = v + mh;
        atomicAdd(&node[(size_t)(row0 + m) * NF + col0 + lm], acc[v]);
    }
}

// ---------------------------------------------------------------------------
// Kernel 2: fused edge -> ConvLSTM1 -> ConvLSTM2 -> adj -> messages.
// 338 blocks x 64 threads (2 waves, one 16-pixel tile each). Weight tiles are
// block-cooperatively staged into LDS (coalesced async b128) and consumed as
// WMMA B fragments from LDS. adj dot-product folded into the h2 epilogue via
// LDS float atomics (no H2 buffer -> fits 64KB static LDS).
// ---------------------------------------------------------------------------
#define LSTR 132   // LDS row stride (floats): rows 16B-aligned, (4r+c)%64 conflict-free
__global__ __launch_bounds__(64) void edge_msg_kernel(
    const float* __restrict__ node,   // (832, 64)
    const float* __restrict__ W1,     // (512, 256) use [:, :128]
    const float* __restrict__ b1,     // (512)
    const float* __restrict__ W2,     // (512, 256) use [:, :128]
    const float* __restrict__ b2,     // (512)
    const float* __restrict__ linkW,  // (128)
    const float* __restrict__ linkb,  // (1)
    const float* __restrict__ mEW,    // (64, 128)
    const float* __restrict__ mEb,    // (64)
    const float* __restrict__ mHW,    // (64, 128)
    const float* __restrict__ mHb,    // (64)
    float* __restrict__ adj,          // (10816)
    float* __restrict__ msum)         // (832, 128) atomic accumulate
{
    __shared__ float Esh[2][16][LSTR];
    __shared__ float H1sh[2][16][LSTR];
    __shared__ float Wl[3][16][LSTR];
    __shared__ float AdjAcc[2][16];
    __shared__ float Adj[2][16];

    const int tid  = threadIdx.x;
    const int wave = tid >> 5;
    const int lane = tid & 31;
    const int tile = blockIdx.x * 2 + wave;   // 0..675
    float (*E)[LSTR]  = Esh[wave];
    float (*H1)[LSTR] = H1sh[wave];

    const int lm = lane & 15;
    const int kh = (lane >> 4) << 1;
    const int mh = (lane >> 4) << 3;

    // ---- Phase A: E[16][128] = concat(node_i, node_w); AdjAcc = 0
    for (int t = lane; t < 16 * 32; t += 32) {   // 512 float4 per wave
        int m = t >> 5;
        int q = t & 31;                          // float4 index in 0..31
        int p  = tile * 16 + m;
        int bt = p / (MAXN * MAXN);
        int r  = p % (MAXN * MAXN);
        int ii = r / MAXN, ww = r % MAXN;
        const float* src = (q < 16)
            ? (node + (size_t)(bt * MAXN + ii) * NF + q * 4)
            : (node + (size_t)(bt * MAXN + ww) * NF + (q - 16) * 4);
        g2l16(src, &E[m][q * 4]);
    }
    if (lane < 16) AdjAcc[wave][lane] = 0.0f;
    g2l_wait();
    __syncthreads();

    // ---- Phase B: ConvLSTM cell 1 (gates i,o,g; f gate dead with zero state)
    for (int nsub = 0; nsub < 8; ++nsub) {
        // stage 3 gate-weight tiles (16 rows x 128 k), block-wide
        for (int t = tid; t < 3 * 16 * 32; t += 64) {   // 1536 float4
            int gate = t >> 9;
            int rem  = t & 511;
            int r    = rem >> 5;
            int q    = (rem & 31) << 2;
            int gb   = (gate == 0) ? 0 : (gate == 1) ? 256 : 384;
            g2l16(W1 + (size_t)(gb + nsub * 16 + r) * 256 + q, &Wl[gate][r][q]);
        }
        g2l_wait();
        __syncthreads();
        v8f ai = {}, ao = {}, ag = {};
        #pragma unroll 4
        for (int kk = 0; kk < 128; kk += 4) {
            v2f a, bi, bo, bg;
            a.x  = E[lm][kk + kh];        a.y  = E[lm][kk + kh + 1];
            bi.x = Wl[0][lm][kk + kh];    bi.y = Wl[0][lm][kk + kh + 1];
            bo.x = Wl[1][lm][kk + kh];    bo.y = Wl[1][lm][kk + kh + 1];
            bg.x = Wl[2][lm][kk + kh];    bg.y = Wl[2][lm][kk + kh + 1];
            ai = wmma4(a, bi, ai);
            ao = wmma4(a, bo, ao);
            ag = wmma4(a, bg, ag);
        }
        float bib = b1[0   + nsub * 16 + lm];
        float bob = b1[256 + nsub * 16 + lm];
        float bgb = b1[384 + nsub * 16 + lm];
        #pragma unroll
        for (int v = 0; v < 8; ++v) {
            float zi = ai[v] + bib, zo = ao[v] + bob, zg = ag[v] + bgb;
            float cc = sigm(zi) * tanhf(zg);
            H1[v + mh][nsub * 16 + lm] = sigm(zo) * tanhf(cc);
        }
        __syncthreads();
    }

    // ---- Phase C: ConvLSTM cell 2; fold link dot-product into epilogue
    for (int nsub = 0; nsub < 8; ++nsub) {
        for (int t = tid; t < 3 * 16 * 32; t += 64) {
            int gate = t >> 9;
            int rem  = t & 511;
            int r    = rem >> 5;
            int q    = (rem & 31) << 2;
            int gb   = (gate == 0) ? 0 : (gate == 1) ? 256 : 384;
            g2l16(W2 + (size_t)(gb + nsub * 16 + r) * 256 + q, &Wl[gate][r][q]);
        }
        g2l_wait();
        __syncthreads();
        v8f ai = {}, ao = {}, ag = {};
        #pragma unroll 4
        for (int kk = 0; kk < 128; kk += 4) {
            v2f a, bi, bo, bg;
            a.x  = H1[lm][kk + kh];       a.y  = H1[lm][kk + kh + 1];
            bi.x = Wl[0][lm][kk + kh];    bi.y = Wl[0][lm][kk + kh + 1];
            bo.x = Wl[1][lm][kk + kh];    bo.y = Wl[1][lm][kk + kh + 1];
            bg.x = Wl[2][lm][kk + kh];    bg.y = Wl[2][lm][kk + kh + 1];
            ai = wmma4(a, bi, ai);
            ao = wmma4(a, bo, ao);
            ag = wmma4(a, bg, ag);
        }
        float bib = b2[0   + nsub * 16 + lm];
        float bob = b2[256 + nsub * 16 + lm];
        float bgb = b2[384 + nsub * 16 + lm];
        float lw  = linkW[nsub * 16 + lm];   // this lane's h2 column weight
        #pragma unroll
        for (int v = 0; v < 8; ++v) {
            float zi = ai[v] + bib, zo = ao[v] + bob, zg = ag[v] + bgb;
            float cc = sigm(zi) * tanhf(zg);
            float h2 = sigm(zo) * tanhf(cc);
            atomicAdd(&AdjAcc[wave][v + mh], lw * h2);   // ds_add_f32
        }
        __syncthreads();
    }

    // ---- Phase D: adjacency = sigmoid(sum + linkb)
    if (lane < 16) {
        float av = sigm(AdjAcc[wave][lane] + linkb[0]);
        adj[tile * 16 + lane] = av;
        Adj[wave][lane] = av;
    }
    __syncthreads();

    // ---- Phase E: messages m = concat(mEW@E+b, mHW@E+b) * adj, sum over w
    for (int nsub = 0; nsub < 8; ++nsub) {
        const float* Wm = (nsub < 4) ? mEW : mHW;
        const float* bm = (nsub < 4) ? mEb : mHb;
        const int rb = (nsub & 3) * 16;
        for (int t = tid; t < 16 * 32; t += 64) {   // 512 float4
            int r = t >> 5;
            int q = (t & 31) << 2;
            g2l16(Wm + (size_t)(rb + r) * 128 + q, &Wl[0][r][q]);
        }
        g2l_wait();
        __syncthreads();
        v8f acc = {};
        #pragma unroll 4
        for (int kk = 0; kk < 128; kk += 4) {
            v2f a, b;
            a.x = E[lm][kk + kh];       a.y = E[lm][kk + kh + 1];
            b.x = Wl[0][lm][kk + kh];   b.y = Wl[0][lm][kk + kh + 1];
            acc = wmma4(a, b, acc);
        }
        const float bb = bm[rb + lm];
        const int ch = nsub * 16 + lm;
        #pragma unroll
        for (int v = 0; v < 8; ++v) {
            int m  = v + mh;
            int p  = tile * 16 + m;
            int bt = p / (MAXN * MAXN);
            int r  = p % (MAXN * MAXN);
            int ii = r / MAXN;
            float val = (acc[v] + bb) * Adj[wave][m];
            atomicAdd(&msum[(size_t)(bt * MAXN + ii) * 128 + ch], val);
        }
        __syncthreads();
    }
}

// ---------------------------------------------------------------------------
// Kernel 3: GRU update + readout -> action_score (832 x 7)
// ---------------------------------------------------------------------------
__global__ __launch_bounds__(64) void gru_readout_kernel(
    const float* __restrict__ msum,  // (832, 128)
    const float* __restrict__ node,  // (832, 64)
    const float* __restrict__ g0ih, const float* __restrict__ g0hh,
    const float* __restrict__ g0bi, const float* __restrict__ g0bh,
    const float* __restrict__ g1ih, const float* __restrict__ g1hh,
    const float* __restrict__ g1bi, const float* __restrict__ g1bh,
    const float* __restrict__ ro0W, const float* __restrict__ ro0b,
    const float* __restrict__ ro1W, const float* __restrict__ ro1b,
    float* __restrict__ out_action)  // (832, 7)
{
    __shared__ float xs[128];
    __shared__ float hs[64];
    __shared__ float hn[64];
    const int nid = blockIdx.x;
    const int i   = nid % MAXN;
    const int c   = threadIdx.x;

    xs[c]      = msum[(size_t)nid * 128 + c];
    xs[c + 64] = msum[(size_t)nid * 128 + 64 + c];
    hs[c]      = node[(size_t)nid * 64 + c];
    __syncthreads();

    const float* Wih = (i == 0) ? g0ih : g1ih;
    const float* Whh = (i == 0) ? g0hh : g1hh;
    const float* bih = (i == 0) ? g0bi : g1bi;
    const float* bhh = (i == 0) ? g0bh : g1bh;

    float gr = bih[c], gz = bih[64 + c], gn = bih[128 + c];
    for (int k = 0; k < 128; ++k) {
        float x = xs[k];
        gr += Wih[(size_t)c * 128 + k] * x;
        gz += Wih[(size_t)(64 + c) * 128 + k] * x;
        gn += Wih[(size_t)(128 + c) * 128 + k] * x;
    }
    float hr = bhh[c], hz = bhh[64 + c], hnn = bhh[128 + c];
    for (int k = 0; k < 64; ++k) {
        float h = hs[k];
        hr  += Whh[(size_t)c * 64 + k] * h;
        hz  += Whh[(size_t)(64 + c) * 64 + k] * h;
        hnn += Whh[(size_t)(128 + c) * 64 + k] * h;
    }
    float r = sigm(gr + hr);
    float z = sigm(gz + hz);
    float n = tanhf(gn + r * hnn);
    hn[c] = (1.0f - z) * n + z * hs[c];
    __syncthreads();

    if (c < C7) {
        const float* roW = (i == 0) ? ro0W : ro1W;
        const float* rob = (i == 0) ? ro0b : ro1b;
        float acc = rob[c];
        for (int k = 0; k < 64; ++k) acc += roW[c * 64 + k] * hn[k];
        out_action[nid * C7 + c] = acc;
    }
}

// ---------------------------------------------------------------------------
// Kernel 4: per-(b,t) softmax over adj (169), zero diagonal -> actor_score;
// weighted row sums -> acty_score.
// ---------------------------------------------------------------------------
__global__ __launch_bounds__(256) void softmax_acty_kernel(
    const float* __restrict__ adj,        // (64, 169)
    const float* __restrict__ action,     // (832, 7)
    float* __restrict__ out_acty,         // (64, 6)
    float* __restrict__ out_actor)        // (64, 169)
{
    __shared__ float vals[169];
    __shared__ float red[256];
    __shared__ float srow[MAXN];
    const int bt  = blockIdx.x;
    const int tid = threadIdx.x;
    const int NN  = MAXN * MAXN;

    float v = (tid < NN) ? adj[bt * NN + tid] : -1e30f;
    red[tid] = v;
    __syncthreads();
    for (int st = 128; st > 0; st >>= 1) {
        if (tid < st) red[tid] = fmaxf(red[tid], red[tid + st]);
        __syncthreads();
    }
    const float mx = red[0];
    __syncthreads();

    float e = (tid < NN) ? __expf(v - mx) : 0.0f;
    red[tid] = e;
    __syncthreads();
    for (int st = 128; st > 0; st >>= 1) {
        if (tid < st) red[tid] += red[tid + st];
        __syncthreads();
    }
    const float denom = red[0];
    __syncthreads();

    if (tid < NN) {
        int i = tid / MAXN, w = tid % MAXN;
        float sm = e / denom;
        if (i == w) sm = 0.0f;
        vals[tid] = sm;
        out_actor[bt * NN + tid] = sm;
    }
    __syncthreads();

    if (tid < MAXN) {
        float acc = 0.0f;
        for (int w = 0; w < MAXN; ++w) acc += vals[tid * MAXN + w];
        srow[tid] = acc;
    }
    __syncthreads();

    if (tid < C7 - 1) {
        int c = tid + 1;
        float acc = 0.0f;
        for (int i = 0; i < MAXN; ++i)
            acc += action[(bt * MAXN + i) * C7 + c] * srow[i];
        out_acty[bt * (C7 - 1) + tid] = acc;
    }
}

// ---------------------------------------------------------------------------
// kernel_launch
// ---------------------------------------------------------------------------
extern "C" void kernel_launch(void* const* d_in, const int* in_sizes, int n_in,
                              void* d_out, int out_size, void* d_ws, size_t ws_size,
                              hipStream_t stream) {
    const float* pose  = (const float*)d_in[0];
    const float* W_fc  = (const float*)d_in[5];
    const float* b_fc  = (const float*)d_in[6];
    const float* W1    = (const float*)d_in[7];
    const float* b1    = (const float*)d_in[8];
    const float* W2    = (const float*)d_in[9];
    const float* b2    = (const float*)d_in[10];
    const float* linkW = (const float*)d_in[11];
    const float* linkb = (const float*)d_in[12];
    const float* mEW   = (const float*)d_in[13];
    const float* mEb   = (const float*)d_in[14];
    const float* mHW   = (const float*)d_in[15];
    const float* mHb   = (const float*)d_in[16];
    const float* g0ih  = (const float*)d_in[17];
    const float* g0hh  = (const float*)d_in[18];
    const float* g0bi  = (const float*)d_in[19];
    const float* g0bh  = (const float*)d_in[20];
    const float* g1ih  = (const float*)d_in[21];
    const float* g1hh  = (const float*)d_in[22];
    const float* g1bi  = (const float*)d_in[23];
    const float* g1bh  = (const float*)d_in[24];
    const float* ro0W  = (const float*)d_in[25];
    const float* ro0b  = (const float*)d_in[26];
    const float* ro1W  = (const float*)d_in[27];
    const float* ro1b  = (const float*)d_in[28];

    float* ws   = (float*)d_ws;
    float* node = ws;                         // 832*64   = 53248
    float* adj  = ws + 53248;                 // 10816
    float* msum = ws + 53248 + 10816;         // 832*128  = 106496

    float* out        = (float*)d_out;
    float* out_action = out;                  // 832*7  = 5824
    float* out_acty   = out + 5824;           // 64*6   = 384
    float* out_actor  = out + 5824 + 384;     // 64*169 = 10816

    init_kernel<<<(NODES * 128 + 255) / 256, 256, 0, stream>>>(node, msum, b_fc);

    dim3 g1(NODES / 16, KSPLIT);
    node_fc_kernel<<<g1, 128, 0, stream>>>(pose, W_fc, node);

    edge_msg_kernel<<<PIX_TILES / 2, 64, 0, stream>>>(
        node, W1, b1, W2, b2, linkW, linkb, mEW, mEb, mHW, mHb, adj, msum);

    gru_readout_kernel<<<NODES, 64, 0, stream>>>(
        msum, node, g0ih, g0hh, g0bi, g0bh, g1ih, g1hh, g1bi, g1bh,
        ro0W, ro0b, ro1W, ro1b, out_action);

    softmax_acty_kernel<<<BT, 256, 0, stream>>>(adj, out_action, out_acty, out_actor);
}